// NPSLoss_72911364817185
// MI455X (gfx1250) — compile-verified
//
#include <hip/hip_runtime.h>

// Problem constants (from setup_inputs: x = [32, 3, 512, 512] fp32, colors = [20,3] fp32)
#define HW      262144      // 512*512 = 2^18
#define CHW     786432      // 3*HW
#define NIMG    32
#define NPIX    8388608     // 32*HW
#define NTILES  524288      // NPIX / 16
#define NCOLORS 20
#define NBLOCKS 2048
#define NTHREADS 256
#define WAVES_TOTAL ((NBLOCKS * NTHREADS) / 32)   // 16384
#define TILES_PER_WAVE (NTILES / WAVES_TOTAL)     // 32

typedef float v2f __attribute__((ext_vector_type(2)));
typedef float v8f __attribute__((ext_vector_type(8)));

// max(a,b) as a single v_med3 (no canonicalize ops, no NaN inputs here)
__device__ __forceinline__ float maxf(float a, float b) {
    return __builtin_amdgcn_fmed3f(a, b, __builtin_inff());
}
// clamp to >= 0 as a single v_med3
__device__ __forceinline__ float relu(float a) {
    return __builtin_amdgcn_fmed3f(a, 0.0f, __builtin_inff());
}

// D = A(16x4) x B(4x16) + 0, full fp32 WMMA.
//   A_j row m = [c_r, c_g, c_b, -0.5*|c|^2] for color (8j + (m&7)), clamped to 19.
//     Rows 0-7 and 8-15 carry the SAME colors, so both 16-lane halves of D see
//     every color in-lane -> no cross-lane reduction needed for the max.
//   B col n  = [p_r, p_g, p_b, 1]^T for pixel n.
//   D[m,n]  = p.c - |c|^2/2 = e  ;  min d2 = |p|^2 - 2*max e
__global__ __launch_bounds__(NTHREADS)
void nps_partial_kernel(const float* __restrict__ x,
                        const float* __restrict__ colors,
                        float* __restrict__ partial) {
    const int lane = threadIdx.x & 31;
    const int lo   = lane & 15;
    const bool hi  = lane >= 16;

    // ---- Build three A-matrix fragments: colors 0-7, 8-15, 16-19(+pad 19) ----
    // ISA A layout (16x4 f32): lanes 0-15: VGPR0=K0, VGPR1=K1 ; lanes 16-31: VGPR0=K2, VGPR1=K3
    v2f A[3];
    #pragma unroll
    for (int j = 0; j < 3; ++j) {
        int c = 8 * j + (lo & 7);
        if (c > NCOLORS - 1) c = NCOLORS - 1;   // pad with color 19 (harmless under max)
        float cr = colors[c * 3 + 0];
        float cg = colors[c * 3 + 1];
        float cb = colors[c * 3 + 2];
        float k  = -0.5f * (cr * cr + cg * cg + cb * cb);
        A[j].x = hi ? cb : cr;
        A[j].y = hi ? k  : cg;
    }

    const int wave = (int)(blockIdx.x * blockDim.x + threadIdx.x) >> 5;
    const int t0   = wave * TILES_PER_WAVE;

    float acc = 0.0f;
    const v8f zeroC = {};

    #pragma unroll 4
    for (int tt = 0; tt < TILES_PER_WAVE; ++tt) {
        const int tile = t0 + tt;
        const int pix  = tile * 16 + lo;        // 16 consecutive pixels per tile
        const int img  = pix >> 18;             // / HW
        const int off  = pix & (HW - 1);
        const float* base = x + img * CHW + off;

        // B fragment (4x16 f32): lanes 0-15: (K0=r, K1=g) ; lanes 16-31: (K2=b, K3=1)
        // Three loads touch exactly 3 unique cachelines (r, g, b) = 12 B/pixel:
        //   rb: low half fetches the r-line, high half the b-line
        //   g : g-line (high half duplicates low-half addresses -> same line)
        //   bl: b-line (already in L0 from rb's high half) -> gives low lanes |p|^2
        float rb = base[hi ? 2 * HW : 0];
        float g  = base[HW];
        float bl = base[2 * HW];
        v2f B;
        B.x = rb;
        B.y = hi ? 1.0f : g;

        v8f D0 = __builtin_amdgcn_wmma_f32_16x16x4_f32(false, A[0], false, B,
                                                       (short)0, zeroC, false, false);
        v8f D1 = __builtin_amdgcn_wmma_f32_16x16x4_f32(false, A[1], false, B,
                                                       (short)0, zeroC, false, false);
        v8f D2 = __builtin_amdgcn_wmma_f32_16x16x4_f32(false, A[2], false, B,
                                                       (short)0, zeroC, false, false);

        // In-lane max over all 20 colors (24 values incl. pad), tree depth ~5.
        float t[8];
        #pragma unroll
        for (int v = 0; v < 8; ++v)
            t[v] = maxf(D0[v], maxf(D1[v], D2[v]));
        float m0 = maxf(t[0], t[1]);
        float m1 = maxf(t[2], t[3]);
        float m2 = maxf(t[4], t[5]);
        float m3 = maxf(t[6], t[7]);
        float m  = maxf(maxf(m0, m1), maxf(m2, m3));

        // low lanes: rb = r, g, bl = b  ->  |p|^2 fully in-lane
        float p2 = __builtin_fmaf(rb, rb, __builtin_fmaf(g, g, bl * bl));
        float d2 = relu(__builtin_fmaf(-2.0f, m, p2));  // min d2, clamped >= 0
        float sd = __builtin_amdgcn_sqrtf(d2);          // raw v_sqrt_f32, ~1 ULP
        acc += hi ? 0.0f : sd;                          // low half owns the 16 pixels
    }

    // ---- deterministic reduction: wave shfl tree -> LDS -> one partial per block ----
    #pragma unroll
    for (int s = 16; s > 0; s >>= 1)
        acc += __shfl_xor(acc, s, 32);

    __shared__ float wsum[NTHREADS / 32];
    const int wid = threadIdx.x >> 5;
    if ((threadIdx.x & 31) == 0) wsum[wid] = acc;
    __syncthreads();
    if (threadIdx.x == 0) {
        float s = 0.0f;
        #pragma unroll
        for (int i = 0; i < NTHREADS / 32; ++i) s += wsum[i];
        partial[blockIdx.x] = s;
    }
}

__global__ __launch_bounds__(NTHREADS)
void nps_finalize_kernel(const float* __restrict__ partial, float* __restrict__ out) {
    __shared__ float sh[NTHREADS];
    float s = 0.0f;
    for (int i = threadIdx.x; i < NBLOCKS; i += NTHREADS) s += partial[i];
    sh[threadIdx.x] = s;
    __syncthreads();
    for (int st = NTHREADS / 2; st > 0; st >>= 1) {
        if ((int)threadIdx.x < st) sh[threadIdx.x] += sh[threadIdx.x + st];
        __syncthreads();
    }
    if (threadIdx.x == 0) out[0] = sh[0] / (float)NPIX;
}

extern "C" void kernel_launch(void* const* d_in, const int* in_sizes, int n_in,
                              void* d_out, int out_size, void* d_ws, size_t ws_size,
                              hipStream_t stream) {
    const float* x      = (const float*)d_in[0];   // [32,3,512,512] fp32
    const float* colors = (const float*)d_in[1];   // [20,3] fp32
    float* partial      = (float*)d_ws;            // NBLOCKS floats, fully rewritten each call
    float* out          = (float*)d_out;           // 1 float

    nps_partial_kernel<<<NBLOCKS, NTHREADS, 0, stream>>>(x, colors, partial);
    nps_finalize_kernel<<<1, NTHREADS, 0, stream>>>(partial, out);
}